// NeuralAdditiveModel_17772574670894
// MI455X (gfx1250) — compile-verified
//
#include <hip/hip_runtime.h>
#include <hip/hip_bf16.h>
#include <math.h>

// Problem constants (from the reference)
#define B_SZ   16384
#define F_SZ   256
#define H1_SZ  32
#define H2_SZ  16
#define FCHUNK 16          // features per wave
#define BTILE  16          // batch rows per WMMA M-tile
#define MT     4           // M-tiles per wave (64 batch rows) -> weight reuse

typedef __attribute__((ext_vector_type(2))) float v2f;
typedef __attribute__((ext_vector_type(8))) float v8f;

// ---------------------------------------------------------------------------
// Kernel 0: zero the logit accumulator (d_out is reused as the accumulator)
// ---------------------------------------------------------------------------
__global__ void nam_zero_kernel(float* __restrict__ acc, int n) {
    int i = blockIdx.x * blockDim.x + threadIdx.x;
    if (i < n) acc[i] = 0.0f;
}

// ---------------------------------------------------------------------------
// Kernel 1: main compute. One wave per (64-row batch tile, 16-feature chunk).
// Per feature: [64x32(h1)] x [32x16(W2)] as 4 M-tiles of 8 chained
// V_WMMA_F32_16X16X4_F32. The 4 M-tile chains share every weight load
// (W1/b1/W2 are batch-independent), so loads/WMMA ~= 1.
// ---------------------------------------------------------------------------
__global__ __launch_bounds__(32) void nam_main_kernel(
    const float* __restrict__ x,    // [B, F]
    const float* __restrict__ W1,   // [F, 1, H1]
    const float* __restrict__ b1,   // [F, H1]
    const float* __restrict__ W2,   // [F, H1, H2]
    const float* __restrict__ b2,   // [F, H2]
    const float* __restrict__ W3,   // [F, H2, 1]
    const float* __restrict__ b3,   // [F, 1]
    float* __restrict__ acc)        // [B] logit accumulator
{
    const int lane = threadIdx.x;              // 0..31 (wave32)
    const int n    = lane & 15;                // WMMA N column / batch row id
    const int h    = lane >> 4;                // wave half (K-split per ISA layout)
    const int b0   = blockIdx.x * (BTILE * MT);
    const int f0   = blockIdx.y * FCHUNK;

    // x row base for each M-tile (row n of each 16-row tile; both halves load same row)
    long xbase[MT];
    #pragma unroll
    for (int m = 0; m < MT; ++m)
        xbase[m] = (long)(b0 + BTILE * m + n) * F_SZ;

    // Weighted-h2 accumulators: racc[m][r] = sum over chunk features of
    // relu(h2pre[row=r+8h of tile m, col=n] + b2[f,n]) * W3[f,n].
    float racc[MT][8];
    #pragma unroll
    for (int m = 0; m < MT; ++m)
        #pragma unroll
        for (int r = 0; r < 8; ++r)
            racc[m][r] = 0.0f;
    float sb3 = 0.0f;                          // sum of b3 over chunk (same for all rows)

    for (int fb = 0; fb < FCHUNK; fb += 4) {
        const int f = f0 + fb;
        // 4 features worth of x per M-tile, one b128 load each
        float xs[MT][4];
        #pragma unroll
        for (int m = 0; m < MT; ++m) {
            const float4 xq = *(const float4*)(x + xbase[m] + f);
            xs[m][0] = xq.x; xs[m][1] = xq.y; xs[m][2] = xq.z; xs[m][3] = xq.w;
        }

        #pragma unroll
        for (int ff = 0; ff < 4; ++ff) {
            const int fi = f + ff;
            const float* w1p = W1 + (long)fi * H1_SZ;
            const float* b1p = b1 + (long)fi * H1_SZ;
            const float* w2p = W2 + (long)fi * (H1_SZ * H2_SZ);

            v8f c[MT];
            #pragma unroll
            for (int m = 0; m < MT; ++m)
                c[m] = (v8f){0.f, 0.f, 0.f, 0.f, 0.f, 0.f, 0.f, 0.f};

            // K = 32 in 8 steps of 4. Per ISA layout:
            //   A vgpr0/vgpr1: K = 4s + 2h, 4s + 2h + 1   (row = lane&15)
            //   B vgpr0/vgpr1: same K rows, column n = lane&15
            #pragma unroll
            for (int s = 0; s < 8; ++s) {
                const int j = 4 * s + 2 * h;
                const float2 w1v = *(const float2*)(w1p + j);   // shared by all M-tiles
                const float2 b1v = *(const float2*)(b1p + j);
                v2f bm;
                bm.x = w2p[(j    ) * H2_SZ + n];                // shared by all M-tiles
                bm.y = w2p[(j + 1) * H2_SZ + n];

                #pragma unroll
                for (int m = 0; m < MT; ++m) {
                    v2f a;
                    a.x = fmaxf(fmaf(xs[m][ff], w1v.x, b1v.x), 0.0f);  // h1[row, j]
                    a.y = fmaxf(fmaf(xs[m][ff], w1v.y, b1v.y), 0.0f);  // h1[row, j+1]
                    // 4 independent accumulation chains share one weight load set
                    c[m] = __builtin_amdgcn_wmma_f32_16x16x4_f32(
                               false, a, false, bm, (short)0, c[m], false, false);
                }
            }

            // ---- Epilogue: h2 = relu(c + b2), weight by W3, accumulate ----
            const float b2v = b2[(long)fi * H2_SZ + n];
            const float w3v = W3[(long)fi * H2_SZ + n];
            sb3 += b3[fi];

            #pragma unroll
            for (int m = 0; m < MT; ++m)
                #pragma unroll
                for (int r = 0; r < 8; ++r)
                    racc[m][r] = fmaf(fmaxf(c[m][r] + b2v, 0.0f), w3v, racc[m][r]);
        }
    }

    // ---- One cross-lane N-reduction per M-tile for the whole chunk ----
    // racc[m][r] lives at (row = r + 8h of tile m, col = n); butterfly over the
    // 16-lane N dimension (xor masks 1,2,4,8 never cross the half boundary).
    #pragma unroll
    for (int m = 0; m < MT; ++m) {
        float rowacc = 0.0f;
        #pragma unroll
        for (int r = 0; r < 8; ++r) {
            float t = racc[m][r];
            t += __shfl_xor(t, 1);
            t += __shfl_xor(t, 2);
            t += __shfl_xor(t, 4);
            t += __shfl_xor(t, 8);
            if (n == r) rowacc = t + sb3;   // chunk's b3 sum, once per row
        }
        // Lanes n<8 of each half own rows b0 + 16m + 8h + n
        if (n < 8) {
            atomicAdd(&acc[b0 + BTILE * m + 8 * h + n], rowacc);
        }
    }
}

// ---------------------------------------------------------------------------
// Kernel 2: in-place sigmoid over the accumulated logits
// ---------------------------------------------------------------------------
__global__ void nam_sigmoid_kernel(float* __restrict__ acc, int n) {
    int i = blockIdx.x * blockDim.x + threadIdx.x;
    if (i < n) {
        float v = acc[i];
        acc[i] = 1.0f / (1.0f + __expf(-v));
    }
}

// ---------------------------------------------------------------------------
extern "C" void kernel_launch(void* const* d_in, const int* in_sizes, int n_in,
                              void* d_out, int out_size, void* d_ws, size_t ws_size,
                              hipStream_t stream) {
    (void)in_sizes; (void)n_in; (void)d_ws; (void)ws_size; (void)out_size;

    const float* x  = (const float*)d_in[0];
    const float* W1 = (const float*)d_in[1];
    const float* b1 = (const float*)d_in[2];
    const float* W2 = (const float*)d_in[3];
    const float* b2 = (const float*)d_in[4];
    const float* W3 = (const float*)d_in[5];
    const float* b3 = (const float*)d_in[6];
    float* out = (float*)d_out;   // [B,1] fp32; used in-place as logit accumulator

    // 1) zero accumulator (deterministic per call; harness poisons buffers)
    nam_zero_kernel<<<(B_SZ + 255) / 256, 256, 0, stream>>>(out, B_SZ);

    // 2) main compute: grid = (batch tiles of 64 rows, feature chunks of 16)
    dim3 grid(B_SZ / (BTILE * MT), F_SZ / FCHUNK);
    nam_main_kernel<<<grid, 32, 0, stream>>>(x, W1, b1, W2, b2, W3, b3, out);

    // 3) sigmoid in place
    nam_sigmoid_kernel<<<(B_SZ + 255) / 256, 256, 0, stream>>>(out, B_SZ);
}